// feature_generation_14439680049266
// MI455X (gfx1250) — compile-verified
//
#include <hip/hip_runtime.h>
#include <hip/hip_bf16.h>

typedef float v2f __attribute__((ext_vector_type(2)));
typedef float v8f __attribute__((ext_vector_type(8)));

#if defined(__gfx1250__) && __has_builtin(__builtin_amdgcn_wmma_f32_16x16x4_f32)
#define HAVE_WMMA_F32 1
#endif

#define NEG_SLOPE 0.1f

__device__ __forceinline__ float leaky(float v) { return v >= 0.f ? v : NEG_SLOPE * v; }

// ---------------------------------------------------------------------------
// Stage 1: per-edge stats: cnt[row] += 1, abssum[row] += |val|
// ---------------------------------------------------------------------------
__global__ void edge_stats_kernel(const long long* __restrict__ row,
                                  const float* __restrict__ val,
                                  float* __restrict__ cnt, float* __restrict__ abssum,
                                  int E) {
    int e = blockIdx.x * blockDim.x + threadIdx.x;
    if (e >= E) return;
    int r = (int)row[e];
    float v = fabsf(val[e]);
    atomicAdd(&cnt[r], 1.f);
    atomicAdd(&abssum[r], v);
}

// ---------------------------------------------------------------------------
// Stage 2: per-node: abs_mean, deg = abssum + abs_mean, dinv = rsqrt(deg),
//          pv_diag = abs_mean * dinv^2, init x1 with diagonal contribution.
// ---------------------------------------------------------------------------
__global__ void node_prep_kernel(const float* __restrict__ cnt,
                                 const float* __restrict__ abssum,
                                 float* __restrict__ dinv, float* __restrict__ pvd,
                                 float* __restrict__ x1, int N) {
    int i = blockIdx.x * blockDim.x + threadIdx.x;
    if (i >= N) return;
    float s = abssum[i];
    float m = s / fmaxf(cnt[i], 1.f);       // abs_mean (scatter_mean of |val|)
    float d = s + m;                        // deg = sum|val| over row + abs_mean (diag)
    float di = d > 0.f ? rsqrtf(d) : 0.f;
    dinv[i] = di;
    float p = m * di * di;                  // pv for the diagonal self-edge
    pvd[i] = p;
    x1[i] = p;                              // conv1 (x=ones): diagonal term of segment_sum
}

// ---------------------------------------------------------------------------
// Stage 3: per-edge pv = val * dinv[row] * dinv[col]; conv1: x1[col] += pv
// ---------------------------------------------------------------------------
__global__ void edge_pv_kernel(const long long* __restrict__ row,
                               const long long* __restrict__ col,
                               const float* __restrict__ val,
                               const float* __restrict__ dinv,
                               float* __restrict__ pv, float* __restrict__ x1, int E) {
    int e = blockIdx.x * blockDim.x + threadIdx.x;
    if (e >= E) return;
    int r = (int)row[e], c = (int)col[e];
    float p = val[e] * dinv[r] * dinv[c];
    pv[e] = p;
    atomicAdd(&x1[c], p);
}

// ---------------------------------------------------------------------------
// Stage 4: rank-1 expand + leaky: X[n,d] = leaky(x1[n]*w1[d] + b1[d]), D=128
// ---------------------------------------------------------------------------
__global__ void expand_kernel(const float* __restrict__ x1, const float* __restrict__ w1,
                              const float* __restrict__ b1, float* __restrict__ X,
                              size_t total) {
    size_t idx = (size_t)blockIdx.x * blockDim.x + threadIdx.x;
    if (idx >= total) return;
    int d = (int)(idx & 127);
    int n = (int)(idx >> 7);
    X[idx] = leaky(x1[n] * w1[d] + b1[d]);
}

// ---------------------------------------------------------------------------
// GEMM: H[n,f] = sum_k X[n,k] * W[f,k] + b[f]   (W row-major (F,K))
// V_WMMA_F32_16X16X4_F32: D(16x16,f32) = A(16x4,f32) x B(4x16,f32) + C
// One 16x16 tile per wave; 8 waves/block share a 16xK LDS stage of X.
//   A frag (ISA 7.12.2, 32-bit A 16x4): lane l -> M = l&15; lanes 0-15 hold
//     K = {0,1} in v[0..1], lanes 16-31 hold K = {2,3}.
//   B frag (mirrored): lane l -> N = l&15; v[0..1] = rows K {0,1} (lanes 0-15)
//     or K {2,3} (lanes 16-31).
//   C/D: VGPR j, lanes 0-15 -> M=j, lanes 16-31 -> M=j+8, col N = l&15.
// ---------------------------------------------------------------------------
__global__ void __launch_bounds__(256)
gemm_wmma_kernel(const float* __restrict__ X, const float* __restrict__ W,
                 const float* __restrict__ bvec, float* __restrict__ H,
                 int Nn, int K, int F) {
    __shared__ float Xs[16 * 256];          // K <= 256 -> 16 KB
    const int tid  = threadIdx.x;
    const int wave = tid >> 5;
    const int lane = tid & 31;
    const int m0 = blockIdx.x * 16;
    const int f0 = blockIdx.y * 128 + wave * 16;

    // stage 16 rows x K floats of X into LDS (float4, rows clamped)
    const int total = 16 * K;
    for (int i = tid * 4; i < total; i += 256 * 4) {
        int r = i / K, k = i - r * K;       // K % 4 == 0 -> float4 stays in-row
        int rr = m0 + r; if (rr >= Nn) rr = Nn - 1;
        *(float4*)&Xs[i] = *(const float4*)&X[(size_t)rr * K + k];
    }
    __syncthreads();
    if (f0 >= F) return;                    // wave-uniform

    const int mn   = lane & 15;             // A-row M / B-col N for this lane
    const int koff = (lane >> 4) * 2;       // K sub-pair: 0 or 2
    const float* wrow = &W[(size_t)(f0 + mn) * K];

    v8f acc = {};
#pragma unroll 4
    for (int kk = 0; kk < K; kk += 4) {
        v2f a, b;
        a.x = Xs[mn * K + kk + koff];
        a.y = Xs[mn * K + kk + koff + 1];
        b.x = wrow[kk + koff];
        b.y = wrow[kk + koff + 1];
#if HAVE_WMMA_F32
        acc = __builtin_amdgcn_wmma_f32_16x16x4_f32(
            /*neg_a=*/false, a, /*neg_b=*/false, b,
            /*c_mod=*/(short)0, acc, /*reuse_a=*/false, /*reuse_b=*/false);
#else
        // scalar fallback (lane owns col N=mn, rows mhalf..mhalf+7)
        {
            int mh = (lane >> 4) * 8;
            for (int j = 0; j < 8; ++j) {
                float s = acc[j];
                for (int k = 0; k < 4; ++k) s += Xs[(mh + j) * K + kk + k] * wrow[kk + k];
                acc[j] = s;
            }
        }
#endif
    }

    const int mhalf = (lane >> 4) * 8;
    const float bv = bvec[f0 + mn];
#pragma unroll
    for (int j = 0; j < 8; ++j) {
        int m = m0 + mhalf + j;
        if (m < Nn) H[(size_t)m * F + f0 + mn] = acc[j] + bv;
    }
}

// ---------------------------------------------------------------------------
// Propagation: out[col[e], :] += pv[e] * h[row[e], :]   (wave per edge)
// ---------------------------------------------------------------------------
__global__ void prop_edges_kernel(const long long* __restrict__ row,
                                  const long long* __restrict__ col,
                                  const float* __restrict__ pv,
                                  const float* __restrict__ h,
                                  float* __restrict__ out, int E, int F) {
    int e = blockIdx.x * (blockDim.x >> 5) + (threadIdx.x >> 5);
    if (e >= E) return;
    int lane = threadIdx.x & 31;
    int r = (int)row[e], c = (int)col[e];
    float p = pv[e];
    const float* hr = &h[(size_t)r * F];
    float* oc = &out[(size_t)c * F];
    for (int f = lane; f < F; f += 32) atomicAdd(&oc[f], p * hr[f]);
}

// Diagonal self-edges: out[i,:] += pvd[i] * h[i,:]
__global__ void prop_diag_kernel(const float* __restrict__ pvd,
                                 const float* __restrict__ h,
                                 float* __restrict__ out, int N, int F) {
    int i = blockIdx.x * (blockDim.x >> 5) + (threadIdx.x >> 5);
    if (i >= N) return;
    int lane = threadIdx.x & 31;
    float p = pvd[i];
    const float* hr = &h[(size_t)i * F];
    float* oc = &out[(size_t)i * F];
    for (int f = lane; f < F; f += 32) atomicAdd(&oc[f], p * hr[f]);
}

// ---------------------------------------------------------------------------
// out = leaky(out + bias)   (F is a power of two)
// ---------------------------------------------------------------------------
__global__ void bias_leaky_kernel(float* __restrict__ x, const float* __restrict__ bias,
                                  size_t total, int fmask) {
    size_t i = (size_t)blockIdx.x * blockDim.x + threadIdx.x;
    if (i >= total) return;
    x[i] = leaky(x[i] + bias[i & fmask]);
}

// ---------------------------------------------------------------------------
extern "C" void kernel_launch(void* const* d_in, const int* in_sizes, int n_in,
                              void* d_out, int out_size, void* d_ws, size_t ws_size,
                              hipStream_t stream) {
    const long long* eidx = (const long long*)d_in[0];   // (2,E) int64
    const float* eval     = (const float*)d_in[1];       // (E,)
    const float* w1 = (const float*)d_in[3];             // (D,1)
    const float* b1 = (const float*)d_in[4];             // (D,)
    const float* w2 = (const float*)d_in[5];             // (2D,D)
    const float* b2 = (const float*)d_in[6];             // (2D,)
    const float* bias2 = (const float*)d_in[7];          // (2D,)
    const float* w3 = (const float*)d_in[8];             // (D,2D)
    const float* b3 = (const float*)d_in[9];             // (D,)
    const float* bias3 = (const float*)d_in[10];         // (D,)
    float* out = (float*)d_out;

    const int E  = in_sizes[1];
    const int D  = in_sizes[4];          // 128
    const int D2 = 2 * D;                // 256
    const int N  = out_size / D;         // 100000

    const long long* row = eidx;
    const long long* col = eidx + E;

    // workspace layout (floats)
    float* ws     = (float*)d_ws;
    float* cnt    = ws;                        // N
    float* abssum = cnt + N;                   // N  (adjacent to cnt: one memset)
    float* dinv   = abssum + N;                // N
    float* pvd    = dinv + N;                  // N
    float* x1     = pvd + N;                   // N
    float* pv     = x1 + N;                    // E
    float* bufA   = pv + E;                    // N*2D  (features ping)
    float* bufB   = bufA + (size_t)N * D2;     // N*2D  (features pong)

    const int T = 256;
    const int egrid = (E + T - 1) / T;
    const int ngrid = (N + T - 1) / T;
    const int wpb = T / 32;                    // waves per block

    // --- normalization (computed once, reused by all 3 convs) ---
    hipMemsetAsync(cnt, 0, (size_t)2 * N * sizeof(float), stream);
    edge_stats_kernel<<<egrid, T, 0, stream>>>(row, eval, cnt, abssum, E);
    node_prep_kernel<<<ngrid, T, 0, stream>>>(cnt, abssum, dinv, pvd, x1, N);
    edge_pv_kernel<<<egrid, T, 0, stream>>>(row, col, eval, dinv, pv, x1, E);

    // --- conv1 result expanded through linear(1->D) + leaky -> bufA (N x D) ---
    expand_kernel<<<(int)(((size_t)N * D + T - 1) / T), T, 0, stream>>>(
        x1, w1, b1, bufA, (size_t)N * D);

    // --- conv2: h2 = bufA @ w2^T + b2 -> bufB (N x 2D) ---
    {
        dim3 grid((N + 15) / 16, D2 / 128);
        gemm_wmma_kernel<<<grid, T, 0, stream>>>(bufA, w2, b2, bufB, N, D, D2);
    }
    hipMemsetAsync(bufA, 0, (size_t)N * D2 * sizeof(float), stream);
    prop_edges_kernel<<<(E + wpb - 1) / wpb, T, 0, stream>>>(row, col, pv, bufB, bufA, E, D2);
    prop_diag_kernel<<<(N + wpb - 1) / wpb, T, 0, stream>>>(pvd, bufB, bufA, N, D2);
    bias_leaky_kernel<<<(int)(((size_t)N * D2 + T - 1) / T), T, 0, stream>>>(
        bufA, bias2, (size_t)N * D2, D2 - 1);

    // --- conv3: h3 = bufA @ w3^T + b3 -> bufB (N x D) ---
    {
        dim3 grid((N + 15) / 16, 1);   // D == 128 -> one y-tile of 8 waves
        gemm_wmma_kernel<<<grid, T, 0, stream>>>(bufA, w3, b3, bufB, N, D2, D);
    }
    hipMemsetAsync(out, 0, (size_t)out_size * sizeof(float), stream);
    prop_edges_kernel<<<(E + wpb - 1) / wpb, T, 0, stream>>>(row, col, pv, bufB, out, E, D);
    prop_diag_kernel<<<(N + wpb - 1) / wpb, T, 0, stream>>>(pvd, bufB, out, N, D);
    bias_leaky_kernel<<<(int)(((size_t)N * D + T - 1) / T), T, 0, stream>>>(
        out, bias3, (size_t)N * D, D - 1);
}